// Net_wxy_17257178595368
// MI455X (gfx1250) — compile-verified
//
#include <hip/hip_runtime.h>
#include <hip/hip_bf16.h>

typedef __attribute__((ext_vector_type(2))) float v2f;
typedef __attribute__((ext_vector_type(8))) float v8f;

#define NN   1024
#define FF   128
#define DD1  64
#define DD2  16
#define EMAX 16384

// ---------------- workspace layout (bytes) ----------------
#define X0_OFF    (size_t)0                     // 1024*64 f32
#define X1_OFF    (X0_OFF   + (size_t)NN*DD1*4)
#define X2_OFF    (X1_OFF   + (size_t)NN*DD1*4)
#define CNT_OFF   (X2_OFF   + (size_t)NN*DD1*4) // 1024 i32
#define CUR_OFF   (CNT_OFF  + (size_t)NN*4)
#define ROW_OFF   (CUR_OFF  + (size_t)NN*4)
#define DINV_OFF  (ROW_OFF  + (size_t)NN*4)
#define NBR_OFF   (DINV_OFF + (size_t)NN*4)     // EMAX i32
#define EIDL_OFF  (NBR_OFF  + (size_t)EMAX*4)
#define EID_OFF   (EIDL_OFF + (size_t)EMAX*4)   // 1024*1024 i32
#define XE_OFF    (EID_OFF  + (size_t)NN*NN*4)  // EMAX*16 f32
#define MUL_OFF   (XE_OFF   + (size_t)EMAX*DD2*4)

// ---------------- setup kernels ----------------
__global__ void k_init(int* __restrict__ eid, int* __restrict__ cnt, int* __restrict__ cur) {
  int g = blockIdx.x * blockDim.x + threadIdx.x;
  if (g < NN * NN) eid[g] = -1;
  if (g < NN) { cnt[g] = 0; cur[g] = 0; }
}

__global__ void k_count(const int* __restrict__ ei, int* __restrict__ cnt,
                        int* __restrict__ eid, int Edir) {
  int e = blockIdx.x * blockDim.x + threadIdx.x;
  if (e >= Edir) return;
  int u = ei[e], v = ei[Edir + e];
  atomicAdd(&cnt[u], 1);
  eid[u * NN + v] = e;
}

__global__ void k_scan(const int* __restrict__ cnt, int* __restrict__ rowstart,
                       float* __restrict__ dinv) {
  __shared__ int tmp[NN];
  int t = threadIdx.x;
  int c = cnt[t];
  tmp[t] = c;
  __syncthreads();
  for (int ofs = 1; ofs < NN; ofs <<= 1) {
    int v = (t >= ofs) ? tmp[t - ofs] : 0;
    __syncthreads();
    tmp[t] += v;
    __syncthreads();
  }
  rowstart[t] = tmp[t] - c;                  // exclusive scan
  dinv[t] = rsqrtf((float)(c + 1));          // degree + self-loop
}

__global__ void k_fill(const int* __restrict__ ei, const int* __restrict__ rowstart,
                       int* __restrict__ cur, int* __restrict__ nbr,
                       int* __restrict__ eidl, int Edir) {
  int e = blockIdx.x * blockDim.x + threadIdx.x;
  if (e >= Edir) return;
  int u = ei[e], v = ei[Edir + e];
  int slot = atomicAdd(&cur[u], 1);
  int p = rowstart[u] + slot;
  nbr[p]  = v;
  eidl[p] = e;
}

// ---------------- GEMM1: x0 = feat(1024x128) @ W1(128x64) + b1, WMMA f32 16x16x4 ----------------
// wave tiling: 64 M-tiles x 4 N-tiles = 256 waves; A frag: lane holds M=lane%16, K=2*(lane/16)+{0,1}
__global__ void k_gemm1(const float* __restrict__ feat, const float* __restrict__ W1,
                        const float* __restrict__ b1, float* __restrict__ x0) {
  int lane = threadIdx.x & 31;
  int wave = (blockIdx.x * blockDim.x + threadIdx.x) >> 5;
  int mt = wave >> 2, nt = wave & 3;
  int r  = lane & 15;
  int kh = lane >> 4;
  int row = mt * 16 + r;
  int col = nt * 16 + r;
  v8f acc = {};
#pragma unroll 4
  for (int k = 0; k < FF / 4; ++k) {
    int kk = 4 * k + 2 * kh;
    v2f a = *reinterpret_cast<const v2f*>(feat + (size_t)row * FF + kk);
    v2f b;
    b.x = W1[(size_t)kk * DD1 + col];
    b.y = W1[(size_t)(kk + 1) * DD1 + col];
    acc = __builtin_amdgcn_wmma_f32_16x16x4_f32(false, a, false, b, (short)0, acc, false, false);
  }
  float bias = b1[col];
#pragma unroll
  for (int j = 0; j < 8; ++j) {
    int m = j + 8 * kh;                       // D layout: VGPR j -> M=j (lanes<16) / M=8+j
    x0[(size_t)(mt * 16 + m) * DD1 + col] = acc[j] + bias;
  }
}

// ---------------- sparse propagation: xout = 0.9*(Ahat @ xin) + 0.1*x0 ----------------
__global__ void k_prop(const float* __restrict__ xin, const float* __restrict__ x0,
                       float* __restrict__ xout, const int* __restrict__ rowstart,
                       const int* __restrict__ cnt, const int* __restrict__ nbr,
                       const float* __restrict__ dinv) {
  int u = blockIdx.x, t = threadIdx.x;       // 64 threads = 64 channels
  float du = dinv[u];
  float acc = du * du * xin[(size_t)u * DD1 + t];
  int start = rowstart[u], len = cnt[u];
  for (int j = 0; j < len; ++j) {
    int v = nbr[start + j];
    acc += du * dinv[v] * xin[(size_t)v * DD1 + t];
  }
  xout[(size_t)u * DD1 + t] = 0.9f * acc + 0.1f * x0[(size_t)u * DD1 + t];
}

// ---------------- edge MLP: xe/mul = gather(x2) (Ex128) @ Wm{1,2}(128x16) + b, WMMA ----------------
__global__ void k_edge(const float* __restrict__ x2, const int* __restrict__ ei,
                       const float* __restrict__ Wm1, const float* __restrict__ bm1,
                       const float* __restrict__ Wm2, const float* __restrict__ bm2,
                       float* __restrict__ xe, float* __restrict__ mul, int Edir) {
  int lane = threadIdx.x & 31;
  int wave = (blockIdx.x * blockDim.x + threadIdx.x) >> 5;
  int nwaves = (Edir + 15) >> 4;
  if (wave >= nwaves) return;
  int r  = lane & 15;
  int kh = lane >> 4;
  int e  = wave * 16 + r;
  int ec = e < Edir ? e : Edir - 1;
  int n0 = ei[ec], n1 = ei[Edir + ec];
  v8f axe = {}, amu = {};
#pragma unroll 2
  for (int k = 0; k < FF / 4; ++k) {
    int kk = 4 * k + 2 * kh;                  // kk even; pair never crosses the 64-boundary
    int node = (kk < DD1) ? n0 : n1;
    v2f a = *reinterpret_cast<const v2f*>(x2 + (size_t)node * DD1 + (kk & (DD1 - 1)));
    v2f b1v, b2v;
    b1v.x = Wm1[(size_t)kk * DD2 + r];       b1v.y = Wm1[(size_t)(kk + 1) * DD2 + r];
    b2v.x = Wm2[(size_t)kk * DD2 + r];       b2v.y = Wm2[(size_t)(kk + 1) * DD2 + r];
    axe = __builtin_amdgcn_wmma_f32_16x16x4_f32(false, a, false, b1v, (short)0, axe, false, false);
    amu = __builtin_amdgcn_wmma_f32_16x16x4_f32(false, a, false, b2v, (short)0, amu, false, false);
  }
  float bb1 = bm1[r], bb2 = bm2[r];
#pragma unroll
  for (int j = 0; j < 8; ++j) {
    int m  = j + 8 * kh;
    int er = wave * 16 + m;
    if (er < Edir) {
      xe[(size_t)er * DD2 + r]  = axe[j] + bb1;
      mul[(size_t)er * DD2 + r] = amu[j] + bb2;
    }
  }
}

// ---------------- per-query: sparse einsum at (a,b),(b,a); W3 matvec; final dot ----------------
__global__ void k_query(const float* __restrict__ x2, const int* __restrict__ posp,
                        const int* __restrict__ rowstart, const int* __restrict__ cnt,
                        const int* __restrict__ nbr, const int* __restrict__ eidl,
                        const int* __restrict__ eid, const float* __restrict__ xe,
                        const float* __restrict__ mul, const float* __restrict__ W3,
                        const float* __restrict__ b3, float* __restrict__ out) {
  __shared__ float sprod[2][16];
  __shared__ float sX[2][16];
  __shared__ float red[64];
  int p = blockIdx.x, t = threadIdx.x;
  int w = t >> 5, lane = t & 31;
  int a = posp[2 * p], b = posp[2 * p + 1];
  int s = (w == 0) ? a : b;                   // ordered pair (s,d)
  int d = (w == 0) ? b : a;
  int start = rowstart[s], len = cnt[s];
  bool anyc = false;
  float pr = 0.f;
  int m = lane & 15;
  for (int j = 0; j < len; ++j) {             // common-neighbor intersection
    int v  = nbr[start + j];
    int e1 = eidl[start + j];
    int e2 = eid[(size_t)v * NN + d];
    if (e2 >= 0) {
      anyc = true;
      pr += xe[(size_t)e1 * DD2 + m] * mul[(size_t)e2 * DD2 + m];
    }
  }
  if (lane < 16) sprod[w][m] = pr;
  __syncthreads();
  float sef = (eid[(size_t)s * NN + d] >= 0) ? 1.f : 0.f;
  bool  U   = anyc || (sef > 0.f);
  if (lane < 16) {
    float y = b3[m] + sef * W3[16 * DD2 + m];
    for (int k = 0; k < 16; ++k) y += sprod[w][k] * W3[k * DD2 + m];
    sX[w][m] = U ? y : 0.f;
  }
  __syncthreads();
  float term = x2[(size_t)a * DD1 + t] * x2[(size_t)b * DD1 + t];  // xx contribution
  if (t < 16) term += sX[0][t] * sX[1][t];                          // ef contribution
  red[t] = term;
  __syncthreads();
  for (int ofs = 32; ofs > 0; ofs >>= 1) {
    if (t < ofs) red[t] += red[t + ofs];
    __syncthreads();
  }
  if (t == 0) out[p] = red[0];
}

// ---------------- launch ----------------
extern "C" void kernel_launch(void* const* d_in, const int* in_sizes, int n_in,
                              void* d_out, int out_size, void* d_ws, size_t ws_size,
                              hipStream_t stream) {
  (void)n_in; (void)ws_size;
  const float* feat = (const float*)d_in[0];
  const float* W1   = (const float*)d_in[1];
  const float* b1   = (const float*)d_in[2];
  const float* Wm1  = (const float*)d_in[3];
  const float* bm1  = (const float*)d_in[4];
  const float* Wm2  = (const float*)d_in[5];
  const float* bm2  = (const float*)d_in[6];
  const float* W3   = (const float*)d_in[7];
  const float* b3   = (const float*)d_in[8];
  const int*   ei   = (const int*)d_in[9];
  const int*   posp = (const int*)d_in[10];
  float*       out  = (float*)d_out;
  int Edir = in_sizes[9] / 2;                 // directed edge count
  int P    = in_sizes[10] / 2;

  char* ws = (char*)d_ws;
  float* x0   = (float*)(ws + X0_OFF);
  float* x1   = (float*)(ws + X1_OFF);
  float* x2   = (float*)(ws + X2_OFF);
  int*   cnt  = (int*)(ws + CNT_OFF);
  int*   cur  = (int*)(ws + CUR_OFF);
  int*   row  = (int*)(ws + ROW_OFF);
  float* dinv = (float*)(ws + DINV_OFF);
  int*   nbr  = (int*)(ws + NBR_OFF);
  int*   eidl = (int*)(ws + EIDL_OFF);
  int*   eid  = (int*)(ws + EID_OFF);
  float* xe   = (float*)(ws + XE_OFF);
  float* mulb = (float*)(ws + MUL_OFF);

  k_init <<<(NN * NN + 255) / 256, 256, 0, stream>>>(eid, cnt, cur);
  k_count<<<(Edir + 255) / 256, 256, 0, stream>>>(ei, cnt, eid, Edir);
  k_scan <<<1, NN, 0, stream>>>(cnt, row, dinv);
  k_fill <<<(Edir + 255) / 256, 256, 0, stream>>>(ei, row, cur, nbr, eidl, Edir);
  k_gemm1<<<32, 256, 0, stream>>>(feat, W1, b1, x0);
  k_prop <<<NN, DD1, 0, stream>>>(x0, x0, x1, row, cnt, nbr, dinv);
  k_prop <<<NN, DD1, 0, stream>>>(x1, x0, x2, row, cnt, nbr, dinv);
  int nwaves = (Edir + 15) / 16;
  k_edge <<<(nwaves + 7) / 8, 256, 0, stream>>>(x2, ei, Wm1, bm1, Wm2, bm2, xe, mulb, Edir);
  k_query<<<P, 64, 0, stream>>>(x2, posp, row, cnt, nbr, eidl, eid, xe, mulb, W3, b3, out);
}